// MolRefiner_9852654977523
// MI455X (gfx1250) — compile-verified
//
#include <hip/hip_runtime.h>
#include <math.h>

typedef float v2f __attribute__((ext_vector_type(2)));
typedef float v8f __attribute__((ext_vector_type(8)));

#define F        64
#define HEADS    4
#define DHEAD    16
#define EDGED    16
#define NLAYERS  4
#define BGR      32
#define NQ       1280
#define MKV      12800
#define NEDGE    5120
#define NPG      40     /* queries per graph  */
#define MPG      400    /* KV atoms per graph */
#define INV_SQRT_D 0.25f

// ---------------------------------------------------------------- utilities
__global__ void copy_x_kernel(const float* __restrict__ x, float* __restrict__ out) {
    int i = blockIdx.x * blockDim.x + threadIdx.x;
    if (i < NQ * F) out[i] = x[i];
}

__global__ void zero_kernel(float* __restrict__ p, int n) {
    int i = blockIdx.x * blockDim.x + threadIdx.x;
    if (i < n) p[i] = 0.f;
}

// ------------------------------------------------ GINE edge message + scatter
// thread = (edge, feature). msg = silu(x[src] + edge_attr @ We); atomic scatter to agg[dst]
__global__ void edge_kernel(const float* __restrict__ xcur,
                            const float* __restrict__ edge_attr,
                            const float* __restrict__ We,   // [16,64] this layer
                            const int*   __restrict__ ei,   // [2,E]
                            float* __restrict__ agg) {
    int t = blockIdx.x * blockDim.x + threadIdx.x;
    if (t >= NEDGE * F) return;
    int e = t >> 6, f = t & 63;
    int s = ei[e], d = ei[NEDGE + e];
    float acc = xcur[s * F + f];
#pragma unroll
    for (int k = 0; k < EDGED; ++k)
        acc = fmaf(edge_attr[e * EDGED + k], We[k * F + f], acc);
    float sig = 1.f / (1.f + expf(-acc));
    atomicAdd(&agg[d * F + f], acc * sig);
}

// ------------------------------------------------ fp32 WMMA GEMM: C = [Cin +] A(Mx64) @ B(64x64)
// grid.x = Mrows/16 (m-tile), 4 waves per block = 4 n-tiles. One 16x16 tile per wave.
__global__ void __launch_bounds__(128)
gemm64_wmma(const float* __restrict__ A, const float* __restrict__ B,
            const float* __restrict__ Cin, float* __restrict__ Cout) {
    const int lane = threadIdx.x & 31;
    const int wave = threadIdx.x >> 5;
    const int mrow = lane & 15;      // M row (A) / N col (B,C)
    const int kh   = lane >> 4;      // which K/M half this lane carries
    const int m0   = blockIdx.x * 16;
    const int n0   = wave * 16;

    v8f c;
    if (Cin) {
#pragma unroll
        for (int r = 0; r < 8; ++r)
            c[r] = Cin[(m0 + r + 8 * kh) * F + n0 + mrow];
    } else {
#pragma unroll
        for (int r = 0; r < 8; ++r) c[r] = 0.f;
    }

    const float* arow = A + (m0 + mrow) * F;
#pragma unroll
    for (int kb = 0; kb < F; kb += 4) {
        const int k0 = kb + 2 * kh;
        v2f a, b;
        a.x = arow[k0 + 0];
        a.y = arow[k0 + 1];
        b.x = B[(k0 + 0) * F + n0 + mrow];
        b.y = B[(k0 + 1) * F + n0 + mrow];
        c = __builtin_amdgcn_wmma_f32_16x16x4_f32(false, a, false, b, (short)0, c,
                                                  false, false);
    }
#pragma unroll
    for (int r = 0; r < 8; ++r)
        Cout[(m0 + r + 8 * kh) * F + n0 + mrow] = c[r];
}

// ------------------------------------------------ per-(graph,head) attention
// S(40x400) = Q Kh^T * 0.25 via WMMA -> LDS; wave-parallel softmax; O = P Vh via WMMA.
__global__ void __launch_bounds__(256)
attn_kernel(const float* __restrict__ Q, const float* __restrict__ Kh,
            const float* __restrict__ Vh, float* __restrict__ O) {
    __shared__ float S[NPG * MPG];          // 40*400*4 = 62.5 KB LDS
    const int g    = blockIdx.x >> 2;       // graph
    const int hh   = blockIdx.x & 3;        // head
    const int lane = threadIdx.x & 31;
    const int wave = threadIdx.x >> 5;      // 0..7
    const int mrow = lane & 15;
    const int kh   = lane >> 4;
    const int qbase  = g * NPG;
    const int kvbase = g * MPG;
    const int hc     = hh * DHEAD;

    // ---- Phase 1: scores. 3 m-tiles (rows padded 40->48) x 25 n-tiles = 75 tiles
    for (int t = wave; t < 75; t += 8) {
        const int mt = t / 25, nt = t % 25;
        const int n0 = nt * 16;
        v8f c;
#pragma unroll
        for (int r = 0; r < 8; ++r) c[r] = 0.f;
        const int qm = mt * 16 + mrow;
#pragma unroll
        for (int kb = 0; kb < DHEAD; kb += 4) {
            const int k0 = kb + 2 * kh;
            v2f a, b;
            a.x = (qm < NPG) ? Q[(qbase + qm) * F + hc + k0 + 0] : 0.f;
            a.y = (qm < NPG) ? Q[(qbase + qm) * F + hc + k0 + 1] : 0.f;
            // B[k][n] = Kh[kv n][k]  (Q . Kh^T)
            b.x = Kh[(kvbase + n0 + mrow) * F + hc + k0 + 0];
            b.y = Kh[(kvbase + n0 + mrow) * F + hc + k0 + 1];
            c = __builtin_amdgcn_wmma_f32_16x16x4_f32(false, a, false, b, (short)0, c,
                                                      false, false);
        }
#pragma unroll
        for (int r = 0; r < 8; ++r) {
            const int m = mt * 16 + r + 8 * kh;
            if (m < NPG) S[m * MPG + n0 + mrow] = c[r] * INV_SQRT_D;
        }
    }
    __syncthreads();

    // ---- Phase 2: softmax, one wave per row (wave32 shuffles)
    for (int row = wave; row < NPG; row += 8) {
        float* sr = S + row * MPG;
        float mx = -INFINITY;
        for (int j = lane; j < MPG; j += 32) mx = fmaxf(mx, sr[j]);
#pragma unroll
        for (int off = 16; off > 0; off >>= 1) mx = fmaxf(mx, __shfl_xor(mx, off, 32));
        float sum = 0.f;
        for (int j = lane; j < MPG; j += 32) { float e = expf(sr[j] - mx); sr[j] = e; sum += e; }
#pragma unroll
        for (int off = 16; off > 0; off >>= 1) sum += __shfl_xor(sum, off, 32);
        const float inv = 1.f / sum;
        for (int j = lane; j < MPG; j += 32) sr[j] *= inv;
    }
    __syncthreads();

    // ---- Phase 3: O = P(40x400) @ Vh(400x16). 3 m-tiles, one per wave 0..2
    if (wave < 3) {
        const int mt = wave;
        v8f c;
#pragma unroll
        for (int r = 0; r < 8; ++r) c[r] = 0.f;
        const int pm = mt * 16 + mrow;
        for (int kb = 0; kb < MPG; kb += 4) {
            const int k0 = kb + 2 * kh;
            v2f a, b;
            a.x = (pm < NPG) ? S[pm * MPG + k0 + 0] : 0.f;
            a.y = (pm < NPG) ? S[pm * MPG + k0 + 1] : 0.f;
            b.x = Vh[(kvbase + k0 + 0) * F + hc + mrow];
            b.y = Vh[(kvbase + k0 + 1) * F + hc + mrow];
            c = __builtin_amdgcn_wmma_f32_16x16x4_f32(false, a, false, b, (short)0, c,
                                                      false, false);
        }
#pragma unroll
        for (int r = 0; r < 8; ++r) {
            const int m = mt * 16 + r + 8 * kh;
            if (m < NPG) O[(qbase + m) * F + hc + mrow] = c[r];
        }
    }
}

// ---------------------------------------------------------------- launcher
extern "C" void kernel_launch(void* const* d_in, const int* in_sizes, int n_in,
                              void* d_out, int out_size, void* d_ws, size_t ws_size,
                              hipStream_t stream) {
    (void)in_sizes; (void)n_in; (void)out_size; (void)ws_size;
    const float* x         = (const float*)d_in[0];
    const float* edge_attr = (const float*)d_in[1];
    const float* Kin       = (const float*)d_in[2];
    const float* Vin       = (const float*)d_in[3];
    const float* We        = (const float*)d_in[4];
    const float* Wm        = (const float*)d_in[5];
    const float* Wq        = (const float*)d_in[6];
    const float* Wk        = (const float*)d_in[7];
    const float* Wv        = (const float*)d_in[8];
    const float* Wo        = (const float*)d_in[9];
    const int*   ei        = (const int*)d_in[10];
    float* xcur = (float*)d_out;   // running residual x

    float* ws  = (float*)d_ws;
    float* agg = ws;               // [N,F]
    float* h   = agg + NQ * F;     // [N,F]
    float* Qb  = h   + NQ * F;     // [N,F]
    float* Ob  = Qb  + NQ * F;     // [N,F]
    float* Khb = Ob  + NQ * F;     // [M,F]
    float* Vhb = Khb + MKV * F;    // [M,F]   total ~7.9 MB

    copy_x_kernel<<<(NQ * F + 255) / 256, 256, 0, stream>>>(x, xcur);

    for (int l = 0; l < NLAYERS; ++l) {
        const float* We_l = We + l * EDGED * F;
        const float* Wm_l = Wm + l * F * F;
        const float* Wq_l = Wq + l * F * F;
        const float* Wk_l = Wk + l * F * F;
        const float* Wv_l = Wv + l * F * F;
        const float* Wo_l = Wo + l * F * F;

        zero_kernel<<<(NQ * F + 255) / 256, 256, 0, stream>>>(agg, NQ * F);
        edge_kernel<<<(NEDGE * F + 255) / 256, 256, 0, stream>>>(xcur, edge_attr, We_l, ei, agg);

        gemm64_wmma<<<NQ / 16, 128, 0, stream>>>(agg, Wm_l, xcur, h);      // h = x + agg@Wm
        gemm64_wmma<<<NQ / 16, 128, 0, stream>>>(h,   Wq_l, nullptr, Qb);  // Q = h@Wq
        gemm64_wmma<<<MKV / 16, 128, 0, stream>>>(Kin, Wk_l, nullptr, Khb);
        gemm64_wmma<<<MKV / 16, 128, 0, stream>>>(Vin, Wv_l, nullptr, Vhb);

        attn_kernel<<<BGR * HEADS, 256, 0, stream>>>(Qb, Khb, Vhb, Ob);

        gemm64_wmma<<<NQ / 16, 128, 0, stream>>>(Ob, Wo_l, xcur, xcur);    // x += o@Wo
    }
}